// PyramidROIAlign_31662498906495
// MI455X (gfx1250) — compile-verified
//
#include <hip/hip_runtime.h>
#include <cmath>

// PyramidROIAlign for MI455X (gfx1250).
// Memory-bound gather+bilinear; WMMA is not applicable (per-pixel weights).
// Strategy: one wave32 per output pixel (box/level math wave-uniform ->
// scalarized), 2x float4 per lane for 8 coalesced b128 gathers + 2 b128
// non-temporal streaming stores per thread. Feature pyramid (178MB) stays
// resident in the 192MB L2; NT stores keep the 100MB output from evicting it.

typedef __attribute__((ext_vector_type(4))) float v4f;

#define B_  2
#define N_  1000
#define C4  64      // 256 channels / 4 floats = 64 float4 chunks per pixel

__device__ __forceinline__ v4f bilerp(v4f tl, v4f tr, v4f bl, v4f br,
                                      float wx, float wy) {
    // Exact reference evaluation order
    v4f top = tl + (tr - tl) * wx;
    v4f bot = bl + (br - bl) * wx;
    return top + (bot - top) * wy;
}

__global__ __launch_bounds__(256) void roi_align_kernel(
    const float* __restrict__ boxes,   // [B,N,4]  (y1,x1,y2,x2)
    const float* __restrict__ meta,    // [B,93]
    const float* __restrict__ f2,      // [B,256,256,256]
    const float* __restrict__ f3,      // [B,128,128,256]
    const float* __restrict__ f4,      // [B, 64, 64,256]
    const float* __restrict__ f5,      // [B, 32, 32,256]
    float* __restrict__ out)           // [B,N,7,7,256]
{
    const int id   = blockIdx.x * 256 + threadIdx.x;
    const int lane = id & 31;          // 32 lanes = one output pixel
    const int pos  = id >> 5;          // box*49 + gy*7 + gx (wave-uniform)
    const int gx   = pos % 7;
    const int t1   = pos / 7;
    const int gy   = t1 % 7;
    const int box  = t1 / 7;           // flat box index 0..B*N-1
    const int b    = box / N_;

    // Box geometry (uniform across the wave)
    const float y1 = boxes[box * 4 + 0];
    const float x1 = boxes[box * 4 + 1];
    const float bh = boxes[box * 4 + 2] - y1;
    const float bw = boxes[box * 4 + 3] - x1;

    // roi_level = clip(4 + round(log2(sqrt(h*w) / (224/sqrt(image_area)))), 2, 5)
    const float area = meta[4] * meta[5];
    const float roi  = log2f(sqrtf(bh * bw) / (224.0f / sqrtf(area)));
    int lvl = 4 + (int)rintf(roi);           // v_rndne: round-half-even == jnp.round
    lvl = lvl < 2 ? 2 : (lvl > 5 ? 5 : lvl);

    const float* fm = (lvl == 2) ? f2 : (lvl == 3) ? f3 : (lvl == 4) ? f4 : f5;
    const int   H   = 256 >> (lvl - 2);
    const int   W   = H;
    const float Hm1 = (float)(H - 1);

    // Sample coordinates (TF crop_and_resize semantics, ph=pw=7)
    const float ys  = y1 * Hm1 + (float)gy * (bh * Hm1 / 6.0f);
    const float xs  = x1 * Hm1 + (float)gx * (bw * Hm1 / 6.0f);
    const float y0f = floorf(ys), x0f = floorf(xs);
    const float wy  = ys - y0f,   wx  = xs - x0f;
    const int y0 = (int)y0f, x0 = (int)x0f;
    const int y0i = min(H - 1, max(0, y0));
    const int yi1 = min(H - 1, max(0, y0 + 1));
    const int x0i = min(W - 1, max(0, x0));
    const int xi1 = min(W - 1, max(0, x0 + 1));

    // Base float4 indices of the four tap pixels
    const v4f* fp  = (const v4f*)fm;
    const int rowb = b * H * W;
    const int i_tl = (rowb + y0i * W + x0i) * C4;
    const int i_tr = (rowb + y0i * W + xi1) * C4;
    const int i_bl = (rowb + yi1 * W + x0i) * C4;
    const int i_br = (rowb + yi1 * W + xi1) * C4;

    // First 512B half: chunks [lane], second 512B half: chunks [32+lane]
    const v4f tl0 = fp[i_tl + lane];       const v4f tl1 = fp[i_tl + 32 + lane];
    const v4f tr0 = fp[i_tr + lane];       const v4f tr1 = fp[i_tr + 32 + lane];
    const v4f bl0 = fp[i_bl + lane];       const v4f bl1 = fp[i_bl + 32 + lane];
    const v4f br0 = fp[i_br + lane];       const v4f br1 = fp[i_br + 32 + lane];

    const v4f r0 = bilerp(tl0, tr0, bl0, br0, wx, wy);
    const v4f r1 = bilerp(tl1, tr1, bl1, br1, wx, wy);

    // Streaming b128 stores (TH_STORE_NT): don't evict the pyramid from L2
    v4f* op = ((v4f*)out) + pos * C4;
    __builtin_nontemporal_store(r0, op + lane);
    __builtin_nontemporal_store(r1, op + 32 + lane);
}

extern "C" void kernel_launch(void* const* d_in, const int* in_sizes, int n_in,
                              void* d_out, int out_size, void* d_ws, size_t ws_size,
                              hipStream_t stream) {
    const float* boxes = (const float*)d_in[0];
    const float* meta  = (const float*)d_in[1];
    const float* f2    = (const float*)d_in[2];
    const float* f3    = (const float*)d_in[3];
    const float* f4    = (const float*)d_in[4];
    const float* f5    = (const float*)d_in[5];
    float* out = (float*)d_out;

    const int total  = B_ * N_ * 7 * 7 * 32;   // 3,136,000 threads
    const int blocks = total / 256;            // 12,250 (exact)
    roi_align_kernel<<<blocks, 256, 0, stream>>>(boxes, meta, f2, f3, f4, f5, out);
}